// Decoder_19129784336708
// MI455X (gfx1250) — compile-verified
//
#include <hip/hip_runtime.h>
#include <hip/hip_bf16.h>

#define BB 32
#define SS 1024
#define EE 256
#define HH 256
#define H4 1024
#define NT 256

typedef __attribute__((ext_vector_type(2))) float v2f;
typedef __attribute__((ext_vector_type(8))) float v8f;

// ---------------------------------------------------------------------------
// fp32 WMMA GEMM: out[M,N] = A[M,K] @ W[K,N] (+ bias[N]).
// V_WMMA_F32_16X16X4_F32, one 16x16 output tile per wave, 8 waves/block.
// N, K are template constants so all strides fold into immediate offsets and
// the K-loop fully unrolls into a back-to-back WMMA chain.
// M is implied by the grid: grid.x * 8 == (M/16)*(N/16) tiles.
// ---------------------------------------------------------------------------
template <int N, int K, int HASBIAS>
__launch_bounds__(256)
__global__ void gemm_wmma_f32(const float* __restrict__ A,    // [M,K]
                              const float* __restrict__ W,    // [K,N]
                              const float* __restrict__ bias, // [N]
                              float* __restrict__ out)        // [M,N]
{
    constexpr int TILES_N = N / 16;
    const int lane    = threadIdx.x & 31;
    const int wave    = threadIdx.x >> 5;
    const int tileIdx = blockIdx.x * 8 + wave;
    const int tm      = tileIdx / TILES_N;
    const int tn      = tileIdx % TILES_N;

    const int l15  = lane & 15;
    const int half = lane >> 4;          // 0 or 1
    const int koff = half * 2;           // K sub-offset per half-wave
    const int m    = tm * 16 + l15;      // A row for this lane
    const int n    = tn * 16 + l15;      // B col for this lane

    const float* Arow = A + (size_t)m * K + koff;
    const float* Wcol = W + (size_t)koff * N + n;

    v8f acc = {};
#pragma unroll
    for (int k = 0; k < K; k += 4) {
        v2f a, b;
        // A 16x4 fp32: v0 = K=0(lanes0-15)/K=2(lanes16-31), v1 = K=1/K=3
        a.x = Arow[k + 0];
        a.y = Arow[k + 1];
        // B 4x16 fp32: rows striped across lanes within a VGPR
        b.x = Wcol[(size_t)(k + 0) * N];
        b.y = Wcol[(size_t)(k + 1) * N];
        acc = __builtin_amdgcn_wmma_f32_16x16x4_f32(
            /*neg_a=*/false, a, /*neg_b=*/false, b,
            /*c_mod=*/(short)0, acc, /*reuse_a=*/false, /*reuse_b=*/false);
    }
    const float bn = HASBIAS ? bias[tn * 16 + l15] : 0.f;
    // C/D layout: VGPR i -> M = i + 8*half, N = l15
#pragma unroll
    for (int i = 0; i < 8; ++i) {
        const int row = tm * 16 + half * 8 + i;
        out[(size_t)row * N + tn * 16 + l15] = acc[i] + bn;
    }
}

// ---------------------------------------------------------------------------
// Persistent pointer-network decoder. One workgroup per batch row (batch rows
// are fully independent: each carries its own x/h/c/mask), 256 threads =
// 8 wave32s. All per-step state lives in LDS; the full 1024-step recurrence
// runs inside the kernel.
//
// If use_pre != 0, the x @ W_ih term is read from the precomputed enc_proj /
// x0_proj tables (4 KB coalesced row fetch) instead of being recomputed as a
// GEMV every step — halving the per-step critical-path MACs and dropping the
// 1 MB/step W_ih re-read from L2.
// ---------------------------------------------------------------------------
__launch_bounds__(NT)
__global__ void ptrnet_decoder(
    const float* __restrict__ enc,     // [B,S,E]
    const float* __restrict__ dec_in,  // [B,E]
    const float* __restrict__ h0,      // [B,H]
    const float* __restrict__ c0,      // [B,H]
    const float* __restrict__ W_ih,    // [E,4H]
    const float* __restrict__ b_ih,    // [4H]
    const float* __restrict__ W_hh,    // [H,4H]
    const float* __restrict__ b_hh,    // [4H]
    const float* __restrict__ W_out,   // [2H,H]
    const float* __restrict__ b_out,   // [H]
    const float* __restrict__ Wa_in,   // [H,H]
    const float* __restrict__ ba_in,   // [H]
    const float* __restrict__ vvec,    // [H]
    const float* __restrict__ ctxp,    // [B,S,H]   ctx_proj (workspace)
    const float* __restrict__ encp,    // [B,S,4H]  enc @ W_ih (workspace)
    const float* __restrict__ x0p,     // [B,4H]    dec_in @ W_ih (workspace)
    int use_pre,
    float* __restrict__ out)           // [B*S*S] alphas ++ [B*S] pointers
{
    const int b    = blockIdx.x;
    const int tid  = threadIdx.x;
    const int lane = tid & 31;
    const int wave = tid >> 5;

    __shared__ float s_x[EE], s_h[HH], s_ht[HH], s_c[HH];
    __shared__ __align__(16) float s_gates[H4];
    __shared__ __align__(16) float s_bg[H4];
    __shared__ float s_inp[HH], s_attn[HH];
    __shared__ float s_scores[SS], s_alpha[SS], s_mask[SS];
    __shared__ float s_v[HH], s_bain[HH], s_bout[HH];
    __shared__ float s_rv[NT];
    __shared__ int   s_ri[NT];
    __shared__ int   s_idx;

    // ---- one-time init ----
    s_x[tid]    = dec_in[b * EE + tid];
    s_h[tid]    = h0[b * HH + tid];
    s_c[tid]    = c0[b * HH + tid];
    s_v[tid]    = vvec[tid];
    s_bain[tid] = ba_in[tid];
    s_bout[tid] = b_out[tid];
    for (int j = tid; j < H4; j += NT) s_bg[j] = b_ih[j] + b_hh[j];
    for (int s = tid; s < SS; s += NT) s_mask[s] = 1.0f;
    __syncthreads();

    const float* ctxb = ctxp + (size_t)b * SS * HH;
    const float* encb = enc  + (size_t)b * SS * EE;
    float* outA = out + (size_t)b * SS * SS;
    float* outP = out + (size_t)BB * SS * SS + (size_t)b * SS;

    int prev_idx = -1;  // pointer chosen at previous step (uniform)

    for (int t = 0; t < SS; ++t) {
        // ---- 1) gates = x@W_ih + h@W_hh + biases (4 columns per thread) ----
        const int j0 = tid * 4;
        float4 acc = *(const float4*)(s_bg + j0);
        if (use_pre) {
            // hoisted x @ W_ih: coalesced 4 KB row fetch
            const float* xp = (t == 0)
                ? (x0p + (size_t)b * H4)
                : (encp + ((size_t)b * SS + prev_idx) * H4);
            const float4 xpr = *(const float4*)(xp + j0);
            acc.x += xpr.x; acc.y += xpr.y; acc.z += xpr.z; acc.w += xpr.w;
        } else {
            for (int e = 0; e < EE; ++e) {
                const float  xe = s_x[e];
                const float4 w  = *(const float4*)(W_ih + (size_t)e * H4 + j0);
                acc.x += xe * w.x; acc.y += xe * w.y;
                acc.z += xe * w.z; acc.w += xe * w.w;
            }
        }
        for (int k = 0; k < HH; ++k) {
            const float  hk = s_h[k];
            const float4 w  = *(const float4*)(W_hh + (size_t)k * H4 + j0);
            acc.x += hk * w.x; acc.y += hk * w.y;
            acc.z += hk * w.z; acc.w += hk * w.w;
        }
        *(float4*)(s_gates + j0) = acc;
        __syncthreads();

        // ---- 2) LSTM cell (order: i, f, g, o) ----
        {
            const float ig = s_gates[tid];
            const float fg = s_gates[HH + tid];
            const float gg = s_gates[2 * HH + tid];
            const float og = s_gates[3 * HH + tid];
            const float si = 1.f / (1.f + __expf(-ig));
            const float sf = 1.f / (1.f + __expf(-fg));
            const float so = 1.f / (1.f + __expf(-og));
            const float ct = sf * s_c[tid] + si * tanhf(gg);
            s_c[tid]  = ct;
            s_ht[tid] = so * tanhf(ct);
        }
        __syncthreads();

        // ---- 3) inp = h_t @ Wa_in + ba_in ----
        {
            float a = s_bain[tid];
            for (int k = 0; k < HH; ++k) a += s_ht[k] * Wa_in[(size_t)k * HH + tid];
            s_inp[tid] = a;
        }
        __syncthreads();

        // ---- 4) scores[s] = v . tanh(inp + ctx_proj[s]) (wave per row) ----
        for (int s = wave; s < SS; s += 8) {
            const float* cp = ctxb + (size_t)s * HH;
            if (s + 8 < SS) __builtin_prefetch(ctxb + (size_t)(s + 8) * HH + lane, 0, 0);
            float a = 0.f;
            for (int h = lane; h < HH; h += 32)
                a += s_v[h] * tanhf(s_inp[h] + cp[h]);
            for (int off = 16; off > 0; off >>= 1)
                a += __shfl_down(a, off, 32);
            if (lane == 0) s_scores[s] = a;
        }
        __syncthreads();

        // ---- 5) masked max + argmax (first-occurrence tie-break) ----
        float mx = -__builtin_inff();
        int   mi = SS;
        for (int q = 0; q < 4; ++q) {
            const int   s  = tid * 4 + q;
            const float sc = (s_mask[s] > 0.f) ? s_scores[s] : -__builtin_inff();
            if (sc > mx) { mx = sc; mi = s; }
        }
        s_rv[tid] = mx; s_ri[tid] = mi;
        __syncthreads();
        for (int stride = NT / 2; stride > 0; stride >>= 1) {
            if (tid < stride) {
                const float ov = s_rv[tid + stride];
                const int   oi = s_ri[tid + stride];
                if (ov > s_rv[tid] || (ov == s_rv[tid] && oi < s_ri[tid])) {
                    s_rv[tid] = ov; s_ri[tid] = oi;
                }
            }
            __syncthreads();
        }
        const float gmax = s_rv[0];
        if (tid == 0) s_idx = s_ri[0];
        __syncthreads();

        // ---- 6) softmax (masked entries -> exp(-inf) = 0) ----
        float ls = 0.f;
        for (int q = 0; q < 4; ++q) {
            const int   s  = tid * 4 + q;
            const float sc = (s_mask[s] > 0.f) ? s_scores[s] : -__builtin_inff();
            const float a  = __expf(sc - gmax);
            s_alpha[s] = a;
            ls += a;
        }
        s_rv[tid] = ls;
        __syncthreads();
        for (int stride = NT / 2; stride > 0; stride >>= 1) {
            if (tid < stride) s_rv[tid] += s_rv[tid + stride];
            __syncthreads();
        }
        const float inv = 1.f / s_rv[0];
        for (int q = 0; q < 4; ++q) {
            const int s = tid * 4 + q;
            s_alpha[s] *= inv;
        }
        __syncthreads();

        // alpha row out (coalesced) + pointer out
        for (int s = tid; s < SS; s += NT) outA[(size_t)t * SS + s] = s_alpha[s];
        if (tid == 0) outP[t] = (float)s_idx;

        // ---- 7) attn_hidden = alpha @ ctx_proj (thread per h, coalesced) ----
        {
            float a = 0.f;
            for (int s = 0; s < SS; ++s) a += s_alpha[s] * ctxb[(size_t)s * HH + tid];
            s_attn[tid] = a;
        }
        __syncthreads();

        // ---- 8) h_out = tanh([attn, h_t] @ W_out + b_out) ----
        {
            float a = s_bout[tid];
            for (int k = 0; k < HH; ++k) a += s_attn[k] * W_out[(size_t)k * HH + tid];
            for (int k = 0; k < HH; ++k) a += s_ht[k] * W_out[(size_t)(HH + k) * HH + tid];
            const float ho = tanhf(a);
            __syncthreads();
            s_h[tid] = ho;
        }

        // ---- 9) pointer/mask/x update ----
        const int idx = s_idx;
        prev_idx = idx;
        if (!use_pre) s_x[tid] = encb[(size_t)idx * EE + tid];
        if (tid == 0) s_mask[idx] = 0.f;
        __syncthreads();
    }
}

// ---------------------------------------------------------------------------
extern "C" void kernel_launch(void* const* d_in, const int* in_sizes, int n_in,
                              void* d_out, int out_size, void* d_ws, size_t ws_size,
                              hipStream_t stream) {
    const float* enc     = (const float*)d_in[0];   // [B,S,E]
    const float* dec_in  = (const float*)d_in[1];   // [B,E]
    const float* h0      = (const float*)d_in[2];   // [B,H]
    const float* c0      = (const float*)d_in[3];   // [B,H]
    const float* context = (const float*)d_in[4];   // [B,S,H]
    const float* W_ih    = (const float*)d_in[5];   // [E,4H]
    const float* b_ih    = (const float*)d_in[6];
    const float* W_hh    = (const float*)d_in[7];   // [H,4H]
    const float* b_hh    = (const float*)d_in[8];
    const float* W_out   = (const float*)d_in[9];   // [2H,H]
    const float* b_out   = (const float*)d_in[10];
    const float* Wa_in   = (const float*)d_in[11];  // [H,H]
    const float* ba_in   = (const float*)d_in[12];
    const float* Wa_ctx  = (const float*)d_in[13];  // [H,H]
    const float* ba_ctx  = (const float*)d_in[14];
    const float* vvec    = (const float*)d_in[15];  // [H]
    float* out = (float*)d_out;

    // Workspace layout (fp32):
    //   ctxp [B*S*H]   = 32 MB   (required)
    //   encp [B*S*4H]  = 128 MB  (optional precompute)
    //   x0p  [B*4H]    = 128 KB  (optional precompute)
    const size_t n_ctx = (size_t)BB * SS * HH;
    const size_t n_enc = (size_t)BB * SS * H4;
    const size_t n_x0  = (size_t)BB * H4;
    float* ctxp = (float*)d_ws;
    float* encp = ctxp + n_ctx;
    float* x0p  = encp + n_enc;
    const int use_pre = (ws_size >= (n_ctx + n_enc + n_x0) * sizeof(float)) ? 1 : 0;

    // ctx_proj = context @ Wa_ctx + ba_ctx  (32768 x 256 x 256)
    // tiles = 2048*16 = 32768, 8 waves/block -> 4096 blocks
    gemm_wmma_f32<HH, HH, 1><<<dim3(4096), dim3(256), 0, stream>>>(
        context, Wa_ctx, ba_ctx, ctxp);

    if (use_pre) {
        // enc_proj = encoder_inputs @ W_ih  (32768 x 1024 x 256)
        // tiles = 2048*64 = 131072 -> 16384 blocks
        gemm_wmma_f32<H4, EE, 0><<<dim3(16384), dim3(256), 0, stream>>>(
            enc, W_ih, nullptr, encp);
        // x0_proj = decoder_input @ W_ih  (32 x 1024 x 256)
        // tiles = 2*64 = 128 -> 16 blocks
        gemm_wmma_f32<H4, EE, 0><<<dim3(16), dim3(256), 0, stream>>>(
            dec_in, W_ih, nullptr, x0p);
    }

    // Persistent decoder: one WGP per batch row, full 1024-step recurrence.
    ptrnet_decoder<<<dim3(BB), dim3(NT), 0, stream>>>(
        enc, dec_in, h0, c0, W_ih, b_ih, W_hh, b_hh,
        W_out, b_out, Wa_in, ba_in, vvec, ctxp, encp, x0p, use_pre, out);
}